// GraphGATConv_46402826666294
// MI455X (gfx1250) — compile-verified
//
#include <hip/hip_runtime.h>
#include <math.h>

#define N_NODES 50000
#define N_EDGES 800000
#define HD 256        // H * D
#define H_HEADS 2
#define D_DIM 128
#define NEG_SLOPE 0.2f
#define LN_EPS 1e-5f

typedef __attribute__((ext_vector_type(2))) float v2f;
typedef __attribute__((ext_vector_type(8))) float v8f;

// ---------- float atomics (lower to global_atomic_add_f32 / max_num_f32) ----------
__device__ __forceinline__ void atomicAddF(float* p, float v) {
    __hip_atomic_fetch_add(p, v, __ATOMIC_RELAXED, __HIP_MEMORY_SCOPE_AGENT);
}
__device__ __forceinline__ void atomicMaxF(float* p, float v) {
    __hip_atomic_fetch_max(p, v, __ATOMIC_RELAXED, __HIP_MEMORY_SCOPE_AGENT);
}

// ================= Kernel 1: z = features @ W via V_WMMA_F32_16X16X4_F32 =======
// Block = 256 threads = 8 waves. Block computes 16 rows x 256 cols of z.
// Each wave owns two 16x16 output tiles (cols [w*32, w*32+32)).
__global__ void __launch_bounds__(256)
gat_gemm_wmma(const float* __restrict__ feat, const float* __restrict__ W,
              float* __restrict__ z) {
    __shared__ float As[16 * 260];               // 16 rows, stride 260 (bank-safe)
    const int tid  = threadIdx.x;
    const int wave = tid >> 5;
    const int lane = tid & 31;
    const int rowBase = blockIdx.x * 16;

    // Stage the 16x256 f32 A-strip into LDS (float4 per thread x4)
    #pragma unroll
    for (int i = 0; i < 4; ++i) {
        int idx = tid + 256 * i;                 // float4 index 0..1023
        int r   = idx >> 6;                      // 64 float4 per row
        int c4  = idx & 63;
        float4 v = ((const float4*)(feat + (size_t)(rowBase + r) * HD))[c4];
        float* dp = &As[r * 260 + c4 * 4];
        dp[0] = v.x; dp[1] = v.y; dp[2] = v.z; dp[3] = v.w;
    }
    __syncthreads();

    const int g  = lane >> 4;                    // lane half: 0 or 1
    const int mn = lane & 15;                    // row (A) / col (B,C)
    const int col0 = wave * 32;

    v8f acc0 = {};
    v8f acc1 = {};

    for (int k0 = 0; k0 < HD; k0 += 4) {
        const int kk = k0 + 2 * g;               // this lane-half's K pair
        // A fragment (ISA 16x4 f32 layout): VGPR0=A[m][2g], VGPR1=A[m][2g+1]
        v2f a;
        a.x = As[mn * 260 + kk];
        a.y = As[mn * 260 + kk + 1];
        // B fragments (4x16): VGPR0 = row kk, VGPR1 = row kk+1, col = mn
        const float* wr0 = W + (size_t)kk * HD;
        const float* wr1 = wr0 + HD;
        __builtin_prefetch(wr0 + 4 * HD, 0, 0);  // global_prefetch_b8 on next K rows
        v2f b0, b1;
        b0.x = wr0[col0 + mn];       b0.y = wr1[col0 + mn];
        b1.x = wr0[col0 + 16 + mn];  b1.y = wr1[col0 + 16 + mn];

        acc0 = __builtin_amdgcn_wmma_f32_16x16x4_f32(false, a, false, b0,
                                                     (short)0, acc0, false, false);
        acc1 = __builtin_amdgcn_wmma_f32_16x16x4_f32(false, a, false, b1,
                                                     (short)0, acc1, false, false);
    }

    // C/D layout: VGPR i, lane half g -> M = i + 8*g, N = mn
    #pragma unroll
    for (int i = 0; i < 8; ++i) {
        const size_t row = (size_t)(rowBase + i + 8 * g) * HD;
        z[row + col0 + mn]      = acc0[i];
        z[row + col0 + 16 + mn] = acc1[i];
    }
}

// ========== Kernel 2: per-(node,head) logits el/er (one wave each) =============
__global__ void __launch_bounds__(256)
gat_logits(const float* __restrict__ z, const float* __restrict__ attn_l,
           const float* __restrict__ attn_r, float* __restrict__ el,
           float* __restrict__ er) {
    const int lane = threadIdx.x & 31;
    const int w = blockIdx.x * 8 + (threadIdx.x >> 5);   // wave id
    if (w >= N_NODES * H_HEADS) return;
    const int n = w >> 1, h = w & 1;
    const float* zp = z + (size_t)n * HD + h * D_DIM;
    const float* al = attn_l + h * D_DIM;
    const float* ar = attn_r + h * D_DIM;
    float sl = 0.f, sr = 0.f;
    #pragma unroll
    for (int t = 0; t < 4; ++t) {
        const int i = lane + 32 * t;
        const float zz = zp[i];
        sl += zz * al[i];
        sr += zz * ar[i];
    }
    #pragma unroll
    for (int off = 16; off > 0; off >>= 1) {
        sl += __shfl_xor(sl, off, 32);
        sr += __shfl_xor(sr, off, 32);
    }
    if (lane == 0) { el[n * 2 + h] = sl; er[n * 2 + h] = sr; }
}

// ========== Kernel 3: reset accumulators (every call; replay-safe) =============
__global__ void __launch_bounds__(256)
gat_init(float* __restrict__ out_agg, float* __restrict__ m,
         float* __restrict__ denom) {
    const int i = blockIdx.x * 256 + threadIdx.x;
    if (i < N_NODES * HD) out_agg[i] = 0.f;
    if (i < N_NODES * H_HEADS) { m[i] = -INFINITY; denom[i] = 0.f; }
}

// ========== Kernel 4: edge scores (leaky relu) + segment max ===================
__global__ void __launch_bounds__(256)
gat_edge_score(const int* __restrict__ src, const int* __restrict__ dst,
               const float* __restrict__ el, const float* __restrict__ er,
               float* __restrict__ sc, float* __restrict__ m) {
    const int e = blockIdx.x * 256 + threadIdx.x;
    if (e >= N_EDGES) return;
    const int s = src[e], d = dst[e];
    #pragma unroll
    for (int h = 0; h < H_HEADS; ++h) {
        float v = el[s * 2 + h] + er[d * 2 + h];
        v = v > 0.f ? v : NEG_SLOPE * v;
        sc[e * 2 + h] = v;
        atomicMaxF(&m[d * 2 + h], v);
    }
}

// ========== Kernel 5: exp(e - m[dst]) + segment sum ============================
__global__ void __launch_bounds__(256)
gat_edge_exp(const int* __restrict__ dst, float* __restrict__ sc,
             const float* __restrict__ m, float* __restrict__ denom) {
    const int e = blockIdx.x * 256 + threadIdx.x;
    if (e >= N_EDGES) return;
    const int d = dst[e];
    #pragma unroll
    for (int h = 0; h < H_HEADS; ++h) {
        const float ex = __expf(sc[e * 2 + h] - m[d * 2 + h]);
        sc[e * 2 + h] = ex;                      // overwrite with exp value
        atomicAddF(&denom[d * 2 + h], ex);
    }
}

// ========== Kernel 6: aggregation out[dst] += alpha * z[src] (wave/edge) =======
__global__ void __launch_bounds__(256)
gat_aggregate(const int* __restrict__ src, const int* __restrict__ dst,
              const float* __restrict__ exs, const float* __restrict__ denom,
              const float* __restrict__ z, float* __restrict__ out_agg) {
    const int lane = threadIdx.x & 31;
    const int e = blockIdx.x * 8 + (threadIdx.x >> 5);   // one wave per edge
    if (e >= N_EDGES) return;
    const int s = src[e], d = dst[e];
    const float a0 = exs[e * 2 + 0] / denom[d * 2 + 0];
    const float a1 = exs[e * 2 + 1] / denom[d * 2 + 1];
    const float* zp = z + (size_t)s * HD;                // L2-resident gather
    float* op = out_agg + (size_t)d * HD;
    #pragma unroll
    for (int t = 0; t < 8; ++t) {
        const int i = lane + 32 * t;                     // coalesced
        const float al = (i < D_DIM) ? a0 : a1;
        atomicAddF(&op[i], al * zp[i]);
    }
}

// ========== Kernel 7: bias + LayerNorm(256) + ELU + head mean (wave/node) ======
__global__ void __launch_bounds__(256)
gat_finalize(const float* __restrict__ out_agg, const float* __restrict__ bias,
             const float* __restrict__ gamma, const float* __restrict__ beta,
             float* __restrict__ out) {
    const int lane = threadIdx.x & 31;
    const int n = blockIdx.x * 8 + (threadIdx.x >> 5);   // one wave per node
    if (n >= N_NODES) return;
    const float* xp = out_agg + (size_t)n * HD;
    float x[8];
    float sum = 0.f, sumsq = 0.f;
    #pragma unroll
    for (int t = 0; t < 8; ++t) {
        const int i = lane + 32 * t;
        x[t] = xp[i] + bias[i];
        sum += x[t];
        sumsq += x[t] * x[t];
    }
    #pragma unroll
    for (int off = 16; off > 0; off >>= 1) {
        sum   += __shfl_xor(sum, off, 32);
        sumsq += __shfl_xor(sumsq, off, 32);
    }
    const float mu   = sum * (1.f / HD);
    const float var  = sumsq * (1.f / HD) - mu * mu;
    const float rstd = rsqrtf(var + LN_EPS);
    #pragma unroll
    for (int t = 0; t < 8; ++t) {
        const int i = lane + 32 * t;
        float y = (x[t] - mu) * rstd * gamma[i] + beta[i];
        x[t] = y > 0.f ? y : expm1f(y);                  // ELU
    }
    // mean over 2 heads: element i pairs with i+128 -> slots t and t+4
    float* op = out + (size_t)n * D_DIM;
    #pragma unroll
    for (int t = 0; t < 4; ++t)
        op[lane + 32 * t] = 0.5f * (x[t] + x[t + 4]);
}

extern "C" void kernel_launch(void* const* d_in, const int* in_sizes, int n_in,
                              void* d_out, int out_size, void* d_ws, size_t ws_size,
                              hipStream_t stream) {
    const float* feat   = (const float*)d_in[0];
    const int*   src    = (const int*)d_in[1];
    const int*   dst    = (const int*)d_in[2];
    const float* W      = (const float*)d_in[3];
    const float* attn_l = (const float*)d_in[4];
    const float* attn_r = (const float*)d_in[5];
    const float* bias   = (const float*)d_in[6];
    const float* gamma  = (const float*)d_in[7];
    const float* beta   = (const float*)d_in[8];
    float* out = (float*)d_out;

    // workspace carve-up (256B aligned)
    char* ws = (char*)d_ws;
    size_t off = 0;
    auto carve = [&](size_t bytes) -> void* {
        void* p = ws + off;
        off += (bytes + 255) & ~(size_t)255;
        return p;
    };
    float* z       = (float*)carve((size_t)N_NODES * HD * 4);       // 51.2 MB
    float* el      = (float*)carve((size_t)N_NODES * H_HEADS * 4);
    float* er      = (float*)carve((size_t)N_NODES * H_HEADS * 4);
    float* m       = (float*)carve((size_t)N_NODES * H_HEADS * 4);
    float* denom   = (float*)carve((size_t)N_NODES * H_HEADS * 4);
    float* exs     = (float*)carve((size_t)N_EDGES * H_HEADS * 4);  // 6.4 MB
    float* out_agg = (float*)carve((size_t)N_NODES * HD * 4);       // 51.2 MB

    // 1) projection GEMM (WMMA f32): 50000/16 = 3125 row-strips
    gat_gemm_wmma<<<N_NODES / 16, 256, 0, stream>>>(feat, W, z);
    // 2) attention logits: one wave per (node, head)
    gat_logits<<<(N_NODES * H_HEADS) / 8, 256, 0, stream>>>(z, attn_l, attn_r, el, er);
    // 3) reset accumulators (every launch — replay safe)
    gat_init<<<(N_NODES * HD + 255) / 256, 256, 0, stream>>>(out_agg, m, denom);
    // 4) edge scores + segment max
    gat_edge_score<<<(N_EDGES + 255) / 256, 256, 0, stream>>>(src, dst, el, er, exs, m);
    // 5) exp + segment sum
    gat_edge_exp<<<(N_EDGES + 255) / 256, 256, 0, stream>>>(dst, exs, m, denom);
    // 6) weighted scatter aggregation: one wave per edge
    gat_aggregate<<<N_EDGES / 8, 256, 0, stream>>>(src, dst, exs, denom, z, out_agg);
    // 7) LayerNorm + ELU + head mean: one wave per node
    gat_finalize<<<N_NODES / 8, 256, 0, stream>>>(out_agg, bias, gamma, beta, out);
}